// GaussianVoxelizer_55722905698855
// MI455X (gfx1250) — compile-verified
//
#include <hip/hip_runtime.h>
#include <hip/hip_bf16.h>
#include <stdint.h>

// ---------------- problem constants (from reference) ----------------
#define DIM_H 200
#define DIM_W 200
#define DIM_D 16
#define NVOX  (DIM_H * DIM_W * DIM_D)     // 640000
#define NROWS (NVOX + 1)                  // 640001  (G = V+1)
#define NCOLS 31                          // 3 means + 1 opa + 18 feats + 9 cov_inv
#define NFEAT 17                          // NUM_CLASSES - 1
#define VOXSZ 0.4f
#define MINX (-40.0f)
#define MINY (-40.0f)
#define MINZ (-1.0f)

typedef __attribute__((ext_vector_type(2))) float v2f;
typedef __attribute__((ext_vector_type(8))) float v8f;

// -------------------------------------------------------------------
// Kernel 0: zero the per-voxel packed keys in workspace
// -------------------------------------------------------------------
__global__ void gv_init_keys(unsigned long long* __restrict__ keys) {
    int g = blockIdx.x * blockDim.x + threadIdx.x;
    if (g < NVOX) keys[g] = 0ull;
}

// -------------------------------------------------------------------
// Kernel 1: scatter-argmax.  key = (opacity_bits << 32) | ~index.
// atomicMax over u64 == segment argmax(opacity) with min-index tiebreak
// (valid because opacity >= 0 so float bits are monotone).
// -------------------------------------------------------------------
__global__ void gv_scatter(const float* __restrict__ means3d,
                           const float* __restrict__ opacities,
                           unsigned long long* __restrict__ keys,
                           int N) {
    int n = blockIdx.x * blockDim.x + threadIdx.x;
    if (n >= N) return;
    float mx = means3d[3 * n + 0];
    float my = means3d[3 * n + 1];
    float mz = means3d[3 * n + 2];
    // jnp.round = round-half-even -> rintf (v_rndne_f32)
    int vx = (int)rintf((mx - MINX) / VOXSZ);
    int vy = (int)rintf((my - MINY) / VOXSZ);
    int vz = (int)rintf((mz - MINZ) / VOXSZ);
    vx = min(max(vx, 0), DIM_H - 1);
    vy = min(max(vy, 0), DIM_W - 1);
    vz = min(max(vz, 0), DIM_D - 1);
    int flat = vx * (DIM_W * DIM_D) + vy * DIM_D + vz;
    unsigned cb = __float_as_uint(opacities[n]);
    unsigned long long key =
        ((unsigned long long)cb << 32) | (unsigned)(~(unsigned)n);
    atomicMax(&keys[flat], key);
}

// -------------------------------------------------------------------
// Kernel 2: per-row gather of means / opacity / features (cols 0..21),
// including the appended "empty" row G-1.
// -------------------------------------------------------------------
__global__ void gv_gather(const unsigned long long* __restrict__ keys,
                          const float* __restrict__ means3d,
                          const float* __restrict__ opacities,
                          const float* __restrict__ features,
                          const float* __restrict__ empty_scalar,
                          float* __restrict__ out,
                          int N) {
    int g = blockIdx.x * blockDim.x + threadIdx.x;
    if (g >= NROWS) return;
    float* row = out + (size_t)g * NCOLS;

    if (g < NVOX) {
        unsigned long long key = keys[g];
        bool valid = (key != 0ull);
        unsigned idx = valid ? (~(unsigned)(key & 0xffffffffu)) : 0u;
        if (idx >= (unsigned)N) idx = (unsigned)N - 1u;  // safety clamp
        float vf = valid ? 1.0f : 0.0f;
        row[0] = means3d[3 * idx + 0] * vf;
        row[1] = means3d[3 * idx + 1] * vf;
        row[2] = means3d[3 * idx + 2] * vf;
        row[3] = opacities[idx] * vf;
#pragma unroll
        for (int c = 0; c < NFEAT; ++c)
            row[4 + c] = features[(size_t)idx * NFEAT + c] * vf;
        row[4 + NFEAT] = 0.0f;  // appended zero class column
    } else {
        // appended background gaussian row
        row[0] = 0.0f; row[1] = 0.0f; row[2] = 2.2f;
        row[3] = 1.0f;
#pragma unroll
        for (int c = 0; c < NFEAT; ++c) row[4 + c] = 0.0f;
        row[4 + NFEAT] = empty_scalar[0];  // EMPTY_LABEL == 17
    }
}

// -------------------------------------------------------------------
// Kernel 3: covariance-inverse via WMMA.
// One wave (32 threads) handles 5 rows.  A (16x4 f32) holds the five
// 3x3 blocks M^T (M[k][i] = s_k * R[k][i]); by symmetry of the ISA
// layouts the identical register pair is also the 4x16 B operand, so
// D = A*B has the five covariances on its diagonal 3x3 blocks.
// D is staged through LDS; lanes 0..4 invert their 3x3 and store
// output columns 22..30.
// -------------------------------------------------------------------
__global__ void gv_cov_inv_wmma(const unsigned long long* __restrict__ keys,
                                const float* __restrict__ scales,
                                const float* __restrict__ rotations,
                                float* __restrict__ out,
                                int N) {
    __shared__ float tile[16 * 16];

    const int lane = threadIdx.x;        // 0..31, single wave32
    const int half = lane >> 4;          // 0 or 1
    const int L    = lane & 15;          // A-matrix row (M index)
    const int vloc = L / 3;              // voxel slot 0..4 (L==15 -> 5, unused)
    const int i    = L % 3;              // column within 3x3 block

    const long long vbase = (long long)blockIdx.x * 5;
    const long long v     = vbase + vloc;           // output row id
    const bool rowActive  = (L < 15) && (v < (long long)NROWS);

    // ---- fetch this row's (scales, quaternion), fully predicated ----
    long long vc = v;
    if (vc >= (long long)NVOX) vc = NVOX - 1;
    if (vc < 0) vc = 0;
    unsigned long long key = keys[vc];
    bool validG = (key != 0ull);
    unsigned idx = validG ? (~(unsigned)(key & 0xffffffffu)) : 0u;
    if (idx >= (unsigned)N) idx = (unsigned)N - 1u;

    float s0 = scales[3 * idx + 0], s1 = scales[3 * idx + 1], s2 = scales[3 * idx + 2];
    float qw = rotations[4 * idx + 0], qx = rotations[4 * idx + 1];
    float qy = rotations[4 * idx + 2], qz = rotations[4 * idx + 3];

    const bool emptyRow = (v == (long long)NVOX);
    const bool useG     = validG && !emptyRow;
    if (!useG) { qw = 1.0f; qx = 0.0f; qy = 0.0f; qz = 0.0f; }
    if (emptyRow)      { s0 = 100.0f; s1 = 100.0f; s2 = 8.0f; }
    else if (!validG)  { s0 = 1.0f;   s1 = 1.0f;   s2 = 1.0f; }

    // ---- R from normalized quaternion ----
    float rn = rsqrtf(qw * qw + qx * qx + qy * qy + qz * qz);
    qw *= rn; qx *= rn; qy *= rn; qz *= rn;
    float R[3][3];
    R[0][0] = 1.0f - 2.0f * (qy * qy + qz * qz);
    R[0][1] = 2.0f * (qx * qy - qw * qz);
    R[0][2] = 2.0f * (qx * qz + qw * qy);
    R[1][0] = 2.0f * (qx * qy + qw * qz);
    R[1][1] = 1.0f - 2.0f * (qx * qx + qz * qz);
    R[1][2] = 2.0f * (qy * qz - qw * qx);
    R[2][0] = 2.0f * (qx * qz - qw * qy);
    R[2][1] = 2.0f * (qy * qz + qw * qx);
    R[2][2] = 1.0f - 2.0f * (qx * qx + qy * qy);
    float sv[3] = { s0, s1, s2 };

    // ---- build the shared A/B operand: lane holds A[L, k0] and A[L, k1] ----
    // 16x4 f32 A layout: VGPR0 = K=0 (lanes 0-15) / K=2 (lanes 16-31),
    //                    VGPR1 = K=1 (lanes 0-15) / K=3 (lanes 16-31).
    int k0 = half ? 2 : 0;
    v2f a;
    a.x = rowActive ? sv[k0] * R[k0][i] : 0.0f;                 // K = 0 or 2
    a.y = (rowActive && !half) ? sv[1] * R[1][i] : 0.0f;        // K = 1 (K=3 is pad)

    // ---- D = A x B (+0): five 3x3 covariances on the block diagonal ----
    v8f c = {};
    v8f d = __builtin_amdgcn_wmma_f32_16x16x4_f32(
        /*neg_a=*/false, a, /*neg_b=*/false, a,
        /*c_mod=*/(short)0, c, /*reuse_a=*/false, /*reuse_b=*/false);

    // ---- spill D (16x16) to LDS: VGPR j = row (8*half + j), col = L ----
#pragma unroll
    for (int j = 0; j < 8; ++j)
        tile[(8 * half + j) * 16 + L] = d[j];
    __syncthreads();

    // ---- lanes 0..4: symmetric 3x3 inverse, store cols 22..30 ----
    if (lane < 5) {
        long long vo = vbase + lane;
        if (vo < (long long)NROWS) {
            int b = 3 * lane;
            float c00 = tile[(b + 0) * 16 + (b + 0)];
            float c01 = tile[(b + 0) * 16 + (b + 1)];
            float c02 = tile[(b + 0) * 16 + (b + 2)];
            float c10 = tile[(b + 1) * 16 + (b + 0)];
            float c11 = tile[(b + 1) * 16 + (b + 1)];
            float c12 = tile[(b + 1) * 16 + (b + 2)];
            float c20 = tile[(b + 2) * 16 + (b + 0)];
            float c21 = tile[(b + 2) * 16 + (b + 1)];
            float c22 = tile[(b + 2) * 16 + (b + 2)];

            float a00 = c11 * c22 - c12 * c21;
            float a01 = c12 * c20 - c10 * c22;
            float a02 = c10 * c21 - c11 * c20;
            float det = c00 * a00 + c01 * a01 + c02 * a02;
            float id  = 1.0f / det;

            float* row = out + (size_t)vo * NCOLS + 22;
            row[0] = a00 * id;
            row[1] = (c02 * c21 - c01 * c22) * id;
            row[2] = (c01 * c12 - c02 * c11) * id;
            row[3] = a01 * id;
            row[4] = (c00 * c22 - c02 * c20) * id;
            row[5] = (c02 * c10 - c00 * c12) * id;
            row[6] = a02 * id;
            row[7] = (c01 * c20 - c00 * c21) * id;
            row[8] = (c00 * c11 - c01 * c10) * id;
        }
    }
}

// -------------------------------------------------------------------
extern "C" void kernel_launch(void* const* d_in, const int* in_sizes, int n_in,
                              void* d_out, int out_size, void* d_ws, size_t ws_size,
                              hipStream_t stream) {
    const float* means3d      = (const float*)d_in[0];
    const float* opacities    = (const float*)d_in[1];
    const float* features     = (const float*)d_in[2];
    const float* scales       = (const float*)d_in[3];
    const float* rotations    = (const float*)d_in[4];
    const float* empty_scalar = (const float*)d_in[5];
    float* out = (float*)d_out;

    const int N = in_sizes[0] / 3;  // number of gaussians

    unsigned long long* keys = (unsigned long long*)d_ws;  // NVOX * 8 bytes

    gv_init_keys<<<(NVOX + 255) / 256, 256, 0, stream>>>(keys);
    gv_scatter<<<(N + 255) / 256, 256, 0, stream>>>(means3d, opacities, keys, N);
    gv_gather<<<(NROWS + 255) / 256, 256, 0, stream>>>(keys, means3d, opacities,
                                                       features, empty_scalar, out, N);
    gv_cov_inv_wmma<<<(NROWS + 4) / 5, 32, 0, stream>>>(keys, scales, rotations, out, N);
}